// UltraLight_VM_UNet_75462575390941
// MI455X (gfx1250) — compile-verified
//
#include <hip/hip_runtime.h>

typedef float v2f __attribute__((ext_vector_type(2)));
typedef float v8f __attribute__((ext_vector_type(8)));

#define DEV __device__ __forceinline__

DEV float geluf(float x){ return 0.5f*x*(1.0f + erff(x*0.7071067811865475f)); }
DEV float siluf(float x){ return x/(1.0f+expf(-x)); }
DEV float softplusf(float x){ return (x>20.0f)?x:log1pf(expf(x)); }

// =================== WMMA f32 GEMM: C[M,N] = A[M,K] @ W[N,K]^T (+bias) ===================
// One wave32 per 16x16 output tile. A/B tiles are staged zero-padded into LDS with
// clamped-address loads + selects (no exec-masked branches), then the K loop is
// branch-free: ds_load_b64 fragments + V_WMMA_F32_16X16X4_F32. K <= 64 everywhere here.
// mode==0: row-major store C[m*N+n]; mode==1: store C[(b*N+n)*Lrows + l] with m=b*Lrows+l.
#define GEMM_KMAX 64
__global__ void k_gemm_wmma(const float* __restrict__ A, const float* __restrict__ W,
                            const float* __restrict__ bias, float* __restrict__ C,
                            int M, int K, int N, int mode, int Lrows)
{
  __shared__ float lA[16 * GEMM_KMAX];
  __shared__ float lB[16 * GEMM_KMAX];
  const int m0 = blockIdx.y * 16, n0 = blockIdx.x * 16;
  const int t  = threadIdx.x;           // 0..31, one wave
  const int Kp = (K + 3) & ~3;          // zero-padded K, multiple of 4

  // ---- stage A (16 x Kp) and B (16 x Kp) into LDS, zero-padded, branch-free ----
  const int Kc = K - 1;
  for (int rr = 0; rr < 16; ++rr) {
    int gm = m0 + rr; int cm = (gm < M) ? gm : (M - 1);
    int gn = n0 + rr; int cn = (gn < N) ? gn : (N - 1);
    for (int kk = t; kk < Kp; kk += 32) {
      int ck = (kk < K) ? kk : Kc;
      float va = A[(long)cm * K + ck];
      float vb = W[(long)cn * K + ck];
      lA[rr * Kp + kk] = (gm < M && kk < K) ? va : 0.0f;
      lB[rr * Kp + kk] = (gn < N && kk < K) ? vb : 0.0f;
    }
  }
  __syncthreads();

  // ---- branch-free WMMA loop from LDS ----
  const int half = t >> 4;
  const int r    = t & 15;
  v8f acc = {0.f,0.f,0.f,0.f,0.f,0.f,0.f,0.f};
  for (int k0 = 0; k0 < Kp; k0 += 4) {
    const int ka = k0 + 2 * half;                 // even -> 8B aligned
    v2f a = *(const v2f*)&lA[r * Kp + ka];
    v2f b = *(const v2f*)&lB[r * Kp + ka];
    acc = __builtin_amdgcn_wmma_f32_16x16x4_f32(false, a, false, b, (short)0, acc, false, false);
  }

  // ---- epilogue (after all WMMAs; divergence here is fine) ----
  const int n = n0 + r;
  if (n >= N) return;
  const float bv = bias ? bias[n] : 0.0f;
#pragma unroll
  for (int i = 0; i < 8; ++i) {
    int m = m0 + i + 8 * half;
    if (m < M) {
      float v = acc[i] + bv;
      if (mode == 0) C[(long)m * N + n] = v;
      else { int bb = m / Lrows, l = m - bb * Lrows; C[((long)bb * N + n) * Lrows + l] = v; }
    }
  }
}

// =================== direct 3x3x3 conv, pad 1 ===================
__global__ void k_conv3d_3(const float* __restrict__ x, const float* __restrict__ w,
                           const float* __restrict__ bias, float* __restrict__ y,
                           int B, int Ci, int Co, int D)
{
  long idx = (long)blockIdx.x * blockDim.x + threadIdx.x;
  long total = (long)B * Co * D * D * D;
  if (idx >= total) return;
  int ww = (int)(idx % D); long r1 = idx / D;
  int hh = (int)(r1 % D);  long r2 = r1 / D;
  int dd = (int)(r2 % D);  long r3 = r2 / D;
  int co = (int)(r3 % Co); int b = (int)(r3 / Co);
  float acc = bias[co];
  for (int ci = 0; ci < Ci; ++ci) {
    const float* xp = x + ((long)(b*Ci + ci) * D) * D * D;
    const float* wp = w + (long)(co*Ci + ci) * 27;
    __builtin_prefetch(xp + ((long)dd*D + hh)*D + ww, 0, 1);
    for (int kd = 0; kd < 3; ++kd) { int zd = dd + kd - 1; if (zd < 0 || zd >= D) continue;
      for (int kh = 0; kh < 3; ++kh) { int zh = hh + kh - 1; if (zh < 0 || zh >= D) continue;
        for (int kw = 0; kw < 3; ++kw) { int zw = ww + kw - 1; if (zw < 0 || zw >= D) continue;
          acc += wp[(kd*3 + kh)*3 + kw] * xp[((long)zd*D + zh)*D + zw];
        } } }
  }
  y[idx] = acc;
}

// =================== 1x1x1 conv, pad 0 ===================
__global__ void k_conv1x1(const float* __restrict__ x, const float* __restrict__ w,
                          const float* __restrict__ bias, float* __restrict__ y,
                          int B, int Ci, int Co, long V)
{
  long idx = (long)blockIdx.x * blockDim.x + threadIdx.x;
  long total = (long)B * Co * V;
  if (idx >= total) return;
  long v = idx % V; long r = idx / V;
  int co = (int)(r % Co); int b = (int)(r / Co);
  float acc = bias[co];
  for (int ci = 0; ci < Ci; ++ci)
    acc += w[co*Ci + ci] * x[((long)(b*Ci + ci))*V + v];
  y[idx] = acc;
}

// =================== 2x2x2 maxpool + GELU ===================
__global__ void k_maxpool_gelu(const float* __restrict__ x, float* __restrict__ y, int NC, int D)
{
  int D2 = D >> 1;
  long idx = (long)blockIdx.x * blockDim.x + threadIdx.x;
  long total = (long)NC * D2 * D2 * D2;
  if (idx >= total) return;
  int ww = (int)(idx % D2); long r1 = idx / D2;
  int hh = (int)(r1 % D2);  long r2 = r1 / D2;
  int dd = (int)(r2 % D2);  int nc = (int)(r2 / D2);
  const float* p = x + (((long)nc * D + 2*dd) * D + 2*hh) * D + 2*ww;
  float m = -1e30f;
  for (int a = 0; a < 2; ++a)
    for (int b2 = 0; b2 < 2; ++b2)
      for (int c = 0; c < 2; ++c) {
        float v = p[((long)a*D + b2)*D + c];
        m = fmaxf(m, v);
      }
  y[idx] = geluf(m);
}

// =================== elementwise ===================
__global__ void k_gelu(float* x, long n){ long i=(long)blockIdx.x*blockDim.x+threadIdx.x; if(i<n) x[i]=geluf(x[i]); }
__global__ void k_sigmoid(float* x, long n){ long i=(long)blockIdx.x*blockDim.x+threadIdx.x; if(i<n) x[i]=1.0f/(1.0f+expf(-x[i])); }
__global__ void k_add(float* x, const float* t, long n){ long i=(long)blockIdx.x*blockDim.x+threadIdx.x; if(i<n) x[i]+=t[i]; }
__global__ void k_fma_gate(float* t, const float* a, long n){ long i=(long)blockIdx.x*blockDim.x+threadIdx.x; if(i<n) t[i] = a[i]*t[i] + t[i]; }

// =================== layernorm (B,C,L)->(B,L,C) ===================
__global__ void k_ln_bcl_blc(const float* __restrict__ x, const float* __restrict__ g,
                             const float* __restrict__ bt, float* __restrict__ y,
                             int B, int C, long L)
{
  long idx = (long)blockIdx.x * blockDim.x + threadIdx.x;
  if (idx >= (long)B * L) return;
  int b = (int)(idx / L); long l = idx % L;
  const float* xp = x + (long)b*C*L + l;
  float m = 0.f;
  for (int c = 0; c < C; ++c) m += xp[(long)c*L];
  m /= (float)C;
  float v = 0.f;
  for (int c = 0; c < C; ++c) { float d = xp[(long)c*L] - m; v += d*d; }
  v /= (float)C;
  float inv = rsqrtf(v + 1e-5f);
  float* yp = y + idx * C;
  for (int c = 0; c < C; ++c) yp[c] = (xp[(long)c*L] - m) * inv * g[c] + bt[c];
}

// =================== layernorm rows (M,C) in-place ===================
__global__ void k_ln_rows(float* __restrict__ x, const float* __restrict__ g,
                          const float* __restrict__ bt, long M, int C)
{
  long idx = (long)blockIdx.x * blockDim.x + threadIdx.x;
  if (idx >= M) return;
  float* xp = x + idx * C;
  float m = 0.f; for (int c = 0; c < C; ++c) m += xp[c];
  m /= (float)C;
  float v = 0.f; for (int c = 0; c < C; ++c) { float d = xp[c] - m; v += d*d; }
  v /= (float)C;
  float inv = rsqrtf(v + 1e-5f);
  for (int c = 0; c < C; ++c) xp[c] = (xp[c] - m) * inv * g[c] + bt[c];
}

// =================== group split / merge for PVM ===================
// xs[((g*B+b)*L + l)*dm + j] = xn[(b*L+l)*C + g*dm + j]
__global__ void k_split4(const float* __restrict__ xn, float* __restrict__ xs, int B, long L, int C)
{
  int dm = C / 4;
  long total = (long)B * L * C;
  long i = (long)blockIdx.x * blockDim.x + threadIdx.x;
  if (i >= total) return;
  int j = (int)(i % dm); long r = i / dm;
  long l = r % L; long r2 = r / L;
  int b = (int)(r2 % B); int g = (int)(r2 / B);
  xs[i] = xn[((long)b*L + l)*C + g*dm + j];
}
__global__ void k_merge4(const float* __restrict__ ymm, const float* __restrict__ xs,
                         const float* __restrict__ skip, float* __restrict__ ym,
                         int B, long L, int C)
{
  int dm = C / 4;
  long total = (long)B * L * C;
  long i = (long)blockIdx.x * blockDim.x + threadIdx.x;
  if (i >= total) return;
  int c = (int)(i % C); long r = i / C;
  long l = r % L; int b = (int)(r / L);
  int g = c / dm, j = c - g*dm;
  long src = (((long)(g*B + b))*L + l)*dm + j;
  ym[i] = ymm[src] + skip[0]*xs[src];
}

// =================== causal depthwise conv (dc=4) + SiLU ===================
// xc is a strided view into xz: row stride 2*di, channel offset 0.
__global__ void k_dwconv_silu(const float* __restrict__ xz, const float* __restrict__ w,
                              const float* __restrict__ b, float* __restrict__ xcs,
                              int Nn, long L, int di)
{
  long total = (long)Nn * L * di;
  long i = (long)blockIdx.x * blockDim.x + threadIdx.x;
  if (i >= total) return;
  int d = (int)(i % di); long r = i / di;
  long l = r % L; int n = (int)(r / L);
  float acc = b[d];
  for (int j = 0; j < 4; ++j) {
    long ls = l - 3 + j;
    if (ls >= 0) acc += w[d*4 + j] * xz[(((long)n*L + ls) * 2 * di) + d];
  }
  xcs[i] = siluf(acc);
}

// =================== dt = softplus(dbl[:, :1] @ dt_w.T + dt_b)  (dtr==1) ===================
__global__ void k_dt(const float* __restrict__ dbl, const float* __restrict__ dtw,
                     const float* __restrict__ dtb, float* __restrict__ dt,
                     long Mr, int di, int nd)
{
  long total = Mr * di;
  long i = (long)blockIdx.x * blockDim.x + threadIdx.x;
  if (i >= total) return;
  int d = (int)(i % di); long m = i / di;
  dt[i] = softplusf(dbl[m*nd] * dtw[d] + dtb[d]);
}

// =================== selective scan, ds=16, fused D*xc + silu(z) gate ===================
__global__ void k_scan(const float* __restrict__ dt, const float* __restrict__ dbl,
                       const float* __restrict__ xcs, const float* __restrict__ xz,
                       const float* __restrict__ A_log, const float* __restrict__ Dp,
                       float* __restrict__ y, int Nn, long L, int di, int nd)
{
  int i = blockIdx.x * blockDim.x + threadIdx.x;
  if (i >= Nn * di) return;
  int d = i % di, n = i / di;
  float An[16], h[16];
#pragma unroll
  for (int s = 0; s < 16; ++s) { An[s] = -expf(A_log[d*16 + s]); h[s] = 0.f; }
  float Dd = Dp[d];
  for (long l = 0; l < L; ++l) {
    long row = (long)n*L + l;
    float dtv = dt[row*di + d];
    float xv  = xcs[row*di + d];
    const float* dr = dbl + row*nd;   // [0]=dt-rank, [1..16]=B, [17..32]=C
    float acc = 0.f;
#pragma unroll
    for (int s = 0; s < 16; ++s) {
      float dA = expf(dtv * An[s]);
      h[s] = dA * h[s] + dtv * dr[1 + s] * xv;
      acc += h[s] * dr[17 + s];
    }
    float zv = xz[row*2*di + di + d];
    y[row*di + d] = (acc + Dd*xv) * siluf(zv);
  }
}

// =================== generic 1-axis linear resize ===================
__global__ void k_resize_axis(const float* __restrict__ x, float* __restrict__ y,
                              long outer, int nin, int nout, long inner, int align)
{
  long total = outer * (long)nout * inner;
  long i = (long)blockIdx.x * blockDim.x + threadIdx.x;
  if (i >= total) return;
  long in_i = i % inner; long r = i / inner;
  int o = (int)(r % nout); long u = r / nout;
  float pos;
  if (align) { float scale = (nout <= 1) ? 0.f : (float)(nin - 1) / (float)(nout - 1); pos = o * scale; }
  else {
    pos = (o + 0.5f) * ((float)nin / (float)nout) - 0.5f;
    pos = fminf(fmaxf(pos, 0.f), (float)(nin - 1));
  }
  int i0 = (int)floorf(pos);
  if (i0 < 0) i0 = 0; if (i0 > nin - 1) i0 = nin - 1;
  int i1 = (i0 + 1 < nin) ? i0 + 1 : nin - 1;
  float wg = pos - (float)i0;
  const float* base = x + (u * (long)nin) * inner + in_i;
  y[i] = base[(long)i0 * inner] * (1.f - wg) + base[(long)i1 * inner] * wg;
}

// =================== axis-max reductions on (BC,1,4,4) ===================
__global__ void k_max3(const float* __restrict__ x, float* mw, float* mh, float* md, int BC)
{
  int i = blockIdx.x * blockDim.x + threadIdx.x;
  if (i >= BC) return;
  const float* p = x + (long)i * 16;
  float all = -1e30f;
  for (int k = 0; k < 16; ++k) all = fmaxf(all, p[k]);
  mw[i] = all; // max over (H,W), D=1
  for (int hh = 0; hh < 4; ++hh) { float m = -1e30f; for (int w2 = 0; w2 < 4; ++w2) m = fmaxf(m, p[hh*4 + w2]); mh[i*4 + hh] = m; }
  for (int w2 = 0; w2 < 4; ++w2) { float m = -1e30f; for (int hh = 0; hh < 4; ++hh) m = fmaxf(m, p[hh*4 + w2]); md[i*4 + w2] = m; }
}

// =================== channel-token attention (tokens = C, feat = dd<=4) + /max norm ===================
__global__ void k_attn(const float* __restrict__ x,
                       const float* qw, const float* qb, const float* kw, const float* kb,
                       const float* vw, const float* vb, float* __restrict__ out,
                       int B, int C, int dd)
{
  int i = blockIdx.x * blockDim.x + threadIdx.x;
  if (i >= B*C) return;
  int b = i / C, iq = i % C;
  float q[4], o[4];
  const float* xr = x + ((long)b*C + iq)*dd;
  for (int oo = 0; oo < dd; ++oo) { float s = qb[oo]; for (int e = 0; e < dd; ++e) s += xr[e]*qw[oo*dd + e]; q[oo] = s; }
  float sc[64];
  float scale = rsqrtf((float)dd);
  float mx = -1e30f;
  for (int j = 0; j < C; ++j) {
    const float* xj = x + ((long)b*C + j)*dd;
    float s = 0.f;
    for (int oo = 0; oo < dd; ++oo) { float kk = kb[oo]; for (int e = 0; e < dd; ++e) kk += xj[e]*kw[oo*dd + e]; s += q[oo]*kk; }
    s *= scale; sc[j] = s; mx = fmaxf(mx, s);
  }
  float den = 0.f;
  for (int j = 0; j < C; ++j) { sc[j] = expf(sc[j] - mx); den += sc[j]; }
  for (int oo = 0; oo < dd; ++oo) o[oo] = 0.f;
  for (int j = 0; j < C; ++j) {
    const float* xj = x + ((long)b*C + j)*dd;
    for (int oo = 0; oo < dd; ++oo) { float vv = vb[oo]; for (int e = 0; e < dd; ++e) vv += xj[e]*vw[oo*dd + e]; o[oo] += sc[j]*vv; }
  }
  float om = -1e30f;
  for (int oo = 0; oo < dd; ++oo) { o[oo] /= den; om = fmaxf(om, o[oo]); }
  for (int oo = 0; oo < dd; ++oo) out[(long)i*dd + oo] = o[oo]/om;
}

// =================== awhd[bc, d_, 0, w_] = aw[bc]*ah[bc*4+d_]*ad[bc*4+w_] ===================
__global__ void k_outer3(const float* aw, const float* ah, const float* ad, float* awhd, int BC)
{
  int i = blockIdx.x * blockDim.x + threadIdx.x;
  if (i >= BC*16) return;
  int w2 = i % 4, d_ = (i / 4) % 4, bc = i / 16;
  awhd[i] = aw[bc] * ah[bc*4 + d_] * ad[bc*4 + w2];
}

// ============================================================================
// host side
// ============================================================================
struct PvmP { const float *ng,*nb,*in_w,*conv_w,*conv_b,*x_w,*dt_w,*dt_b,*A_log,*D,*out_w,*pw,*pb,*skip; };
struct AttP { const float *qw,*qb,*kw,*kb,*vw,*vb; };

#define GRID1D(n) dim3((unsigned)(((n) + 255) / 256)), dim3(256), 0, st

static void gemm(hipStream_t st, const float* A, const float* W, const float* bias, float* C,
                 long M, int K, int N, int mode, int Lrows)
{
  dim3 grid((unsigned)((N + 15) / 16), (unsigned)((M + 15) / 16));
  k_gemm_wmma<<<grid, dim3(32), 0, st>>>(A, W, bias, C, (int)M, K, N, mode, Lrows);
}

static void resize_axis(hipStream_t st, const float* in, float* out, long outer, int nin, int nout, long inner, int align)
{
  long n = outer * (long)nout * inner;
  k_resize_axis<<<GRID1D(n)>>>(in, out, outer, nin, nout, inner, align);
}

// full 3-pass resize (D,H,W)->(d2,h2,w2); tmpA/tmpB scratch
static void resize3(hipStream_t st, const float* in, float* out, long BC,
                    int D, int H, int W, int d2, int h2, int w2, int align,
                    float* tmpA, float* tmpB)
{
  resize_axis(st, in,   tmpA, BC,                 D, d2, (long)H * W, align);
  resize_axis(st, tmpA, tmpB, BC * (long)d2,      H, h2, (long)W,     align);
  resize_axis(st, tmpB, out,  BC * (long)d2 * h2, W, w2, 1,           align);
}

static void run_pvm(hipStream_t st, const float* x, float* out, const PvmP& p,
                    int B, int C, int dout, int S, float* scr)
{
  long L = (long)S * S * S;
  int dm = C / 4, di = 2 * dm;
  const int nd = 1 + 32;          // dt-rank(1) + 2*ds(16)
  long Mr = (long)4 * B * L;
  float* xn  = scr;
  float* xs  = xn  + (long)B * L * C;
  float* xz  = xs  + (long)B * L * C;
  float* xcs = xz  + Mr * 2 * di;
  float* dbl = xcs + Mr * di;
  float* dt  = dbl + Mr * nd;
  float* y   = dt  + Mr * di;
  float* ymm = y   + Mr * di;
  float* ym  = ymm + Mr * dm;

  long BLC = (long)B * L * C;
  k_ln_bcl_blc<<<GRID1D((long)B*L)>>>(x, p.ng, p.nb, xn, B, C, L);
  k_split4<<<GRID1D(BLC)>>>(xn, xs, B, L, C);
  gemm(st, xs, p.in_w, nullptr, xz, Mr, dm, 2*di, 0, 0);              // in_proj
  k_dwconv_silu<<<GRID1D(Mr*(long)di)>>>(xz, p.conv_w, p.conv_b, xcs, 4*B, L, di);
  gemm(st, xcs, p.x_w, nullptr, dbl, Mr, di, nd, 0, 0);               // x_proj
  k_dt<<<GRID1D(Mr*(long)di)>>>(dbl, p.dt_w, p.dt_b, dt, Mr, di, nd);
  k_scan<<<GRID1D((long)4*B*di)>>>(dt, dbl, xcs, xz, p.A_log, p.D, y, 4*B, L, di, nd);
  gemm(st, y, p.out_w, nullptr, ymm, Mr, di, dm, 0, 0);               // out_proj
  k_merge4<<<GRID1D(BLC)>>>(ymm, xs, p.skip, ym, B, L, C);
  k_ln_rows<<<GRID1D((long)B*L)>>>(ym, p.ng, p.nb, (long)B*L, C);
  gemm(st, ym, p.pw, p.pb, out, (long)B*L, C, dout, 1, (int)L);       // projection -> (B,dout,L)
}

static void run_bridge_one(hipStream_t st, float* t, int B, int C, int S,
                           const AttP& pw, const AttP& ph, const AttP& pd,
                           float* scr, float* tmp1, float* tmp2)
{
  int BC = B * C;
  float* r1   = scr;
  float* r2   = r1 + (long)BC * S * S;
  float* r3   = r2 + (long)BC * 4 * S;
  float* mw   = r3 + (long)BC * 16;
  float* mh   = mw + BC;
  float* md   = mh + BC*4;
  float* aw   = md + BC*4;
  float* ah   = aw + BC;
  float* ad   = ah + BC*4;
  float* awhd = ad + BC*4;

  // resize t (S,S,S) -> (1,4,4), half-pixel
  resize_axis(st, t,  r1, BC,          S, 1, (long)S*S, 0);
  resize_axis(st, r1, r2, BC,          S, 4, S,         0);
  resize_axis(st, r2, r3, (long)BC*4,  S, 4, 1,         0);
  k_max3<<<GRID1D(BC)>>>(r3, mw, mh, md, BC);
  k_attn<<<GRID1D(BC)>>>(mw, pw.qw, pw.qb, pw.kw, pw.kb, pw.vw, pw.vb, aw, B, C, 1);
  k_attn<<<GRID1D(BC)>>>(mh, ph.qw, ph.qb, ph.kw, ph.kb, ph.vw, ph.vb, ah, B, C, 4);
  k_attn<<<GRID1D(BC)>>>(md, pd.qw, pd.qb, pd.kw, pd.kb, pd.vw, pd.vb, ad, B, C, 4);
  k_outer3<<<GRID1D(BC*16)>>>(aw, ah, ad, awhd, BC);
  // resize awhd (4,1,4) -> (S,S,S), half-pixel
  resize_axis(st, awhd, tmp2, BC,            4, S, 4, 0);  // (BC,S,1,4)
  resize_axis(st, tmp2, tmp1, (long)BC*S,    1, S, 4, 0);  // (BC,S,S,4)
  resize_axis(st, tmp1, tmp2, (long)BC*S*S,  4, S, 1, 0);  // (BC,S,S,S)
  k_fma_gate<<<GRID1D((long)BC*S*S*S)>>>(t, tmp2, (long)BC*S*S*S);
}

extern "C" void kernel_launch(void* const* d_in, const int* in_sizes, int n_in,
                              void* d_out, int out_size, void* d_ws, size_t ws_size,
                              hipStream_t stream)
{
  hipStream_t st = stream;
  (void)in_sizes; (void)n_in; (void)out_size; (void)ws_size;

  // ---- parse inputs in setup_inputs() dict order ----
  int ip = 0;
  auto nxt = [&]() { return (const float*)d_in[ip++]; };
  const float* x = nxt();
  struct ConvP { const float *w, *b; };
  auto conv_p = [&]() { ConvP c; c.w = nxt(); c.b = nxt(); return c; };
  auto pvm_p  = [&]() { PvmP p; p.ng=nxt(); p.nb=nxt(); p.in_w=nxt(); p.conv_w=nxt(); p.conv_b=nxt();
                        p.x_w=nxt(); p.dt_w=nxt(); p.dt_b=nxt(); p.A_log=nxt(); p.D=nxt();
                        p.out_w=nxt(); p.pw=nxt(); p.pb=nxt(); p.skip=nxt(); return p; };
  auto att_p  = [&]() { AttP a; a.qw=nxt(); a.qb=nxt(); a.kw=nxt(); a.kb=nxt(); a.vw=nxt(); a.vb=nxt(); return a; };

  ConvP enc1 = conv_p(), enc2 = conv_p(), enc3 = conv_p();
  PvmP  enc4 = pvm_p(), enc5 = pvm_p(), enc6 = pvm_p();
  PvmP  dec1 = pvm_p(), dec2 = pvm_p(), dec3 = pvm_p();
  ConvP dec4 = conv_p(), dec5 = conv_p(), fin = conv_p();
  AttP  attw = att_p(), atth = att_p(), attd = att_p();

  // ---- workspace plan (floats) ----
  float* Wsp = (float*)d_ws;
  size_t off = 0;
  auto alloc = [&](size_t n) { float* p = Wsp + off; off += n; return p; };
  float* bufA = alloc(14155776);   // largest conv output (2,8,96^3)
  float* bufB = alloc(2200000);
  float* tmp1 = alloc(2000000);
  float* tmp2 = alloc(2000000);
  float* t1   = alloc(1769472);    // (2,8,48^3)
  float* t2   = alloc(442368);     // (2,16,24^3)
  float* t3   = alloc(82944);      // (2,24,12^3)
  float* t4   = alloc(13824);      // (2,32,6^3)
  float* t5   = alloc(2592);       // (2,48,3^3)
  float* pscr = alloc(2200000);    // pvm scratch
  float* bscr = alloc(600000);     // bridge scratch

  const int B = 2;

  // ---- encoder ----
  k_conv3d_3<<<GRID1D((long)B*8*96*96*96)>>>(x,  enc1.w, enc1.b, bufA, B, 3, 8, 96);
  k_maxpool_gelu<<<GRID1D((long)B*8*48*48*48)>>>(bufA, t1, B*8, 96);
  k_conv3d_3<<<GRID1D((long)B*16*48*48*48)>>>(t1, enc2.w, enc2.b, bufA, B, 8, 16, 48);
  k_maxpool_gelu<<<GRID1D((long)B*16*24*24*24)>>>(bufA, t2, B*16, 48);
  k_conv3d_3<<<GRID1D((long)B*24*24*24*24)>>>(t2, enc3.w, enc3.b, bufA, B, 16, 24, 24);
  k_maxpool_gelu<<<GRID1D((long)B*24*12*12*12)>>>(bufA, t3, B*24, 24);
  run_pvm(st, t3, bufA, enc4, B, 24, 32, 12, pscr);
  k_maxpool_gelu<<<GRID1D((long)B*32*6*6*6)>>>(bufA, t4, B*32, 12);
  run_pvm(st, t4, bufA, enc5, B, 32, 48, 6, pscr);
  k_maxpool_gelu<<<GRID1D((long)B*48*3*3*3)>>>(bufA, t5, B*48, 6);

  // enc6 consumes pre-bridge t5; pvm does not modify its input
  run_pvm(st, t5, bufA, enc6, B, 48, 64, 3, pscr);   // bufA = (2,64,3^3)

  // ---- bridge (modifies t1..t5 in place) ----
  run_bridge_one(st, t1, B,  8, 48, attw, atth, attd, bscr, tmp1, tmp2);
  run_bridge_one(st, t2, B, 16, 24, attw, atth, attd, bscr, tmp1, tmp2);
  run_bridge_one(st, t3, B, 24, 12, attw, atth, attd, bscr, tmp1, tmp2);
  run_bridge_one(st, t4, B, 32,  6, attw, atth, attd, bscr, tmp1, tmp2);
  run_bridge_one(st, t5, B, 48,  3, attw, atth, attd, bscr, tmp1, tmp2);

  // ---- decoder ----
  k_gelu<<<GRID1D((long)B*64*27)>>>(bufA, (long)B*64*27);

  run_pvm(st, bufA, bufB, dec1, B, 64, 48, 3, pscr);
  k_gelu<<<GRID1D((long)B*48*27)>>>(bufB, (long)B*48*27);
  k_add<<<GRID1D((long)B*48*27)>>>(bufB, t5, (long)B*48*27);

  run_pvm(st, bufB, bufA, dec2, B, 48, 32, 3, pscr);
  resize3(st, bufA, bufB, (long)B*32, 3,3,3, 6,6,6, 1, tmp1, tmp2);
  k_gelu<<<GRID1D((long)B*32*216)>>>(bufB, (long)B*32*216);
  k_add<<<GRID1D((long)B*32*216)>>>(bufB, t4, (long)B*32*216);

  run_pvm(st, bufB, bufA, dec3, B, 32, 24, 6, pscr);
  resize3(st, bufA, bufB, (long)B*24, 6,6,6, 12,12,12, 1, tmp1, tmp2);
  k_gelu<<<GRID1D((long)B*24*1728)>>>(bufB, (long)B*24*1728);
  k_add<<<GRID1D((long)B*24*1728)>>>(bufB, t3, (long)B*24*1728);

  k_conv3d_3<<<GRID1D((long)B*16*12*12*12)>>>(bufB, dec4.w, dec4.b, bufA, B, 24, 16, 12);
  resize3(st, bufA, bufB, (long)B*16, 12,12,12, 24,24,24, 1, tmp1, tmp2);
  k_gelu<<<GRID1D((long)B*16*13824)>>>(bufB, (long)B*16*13824);
  k_add<<<GRID1D((long)B*16*13824)>>>(bufB, t2, (long)B*16*13824);

  k_conv3d_3<<<GRID1D((long)B*8*24*24*24)>>>(bufB, dec5.w, dec5.b, bufA, B, 16, 8, 24);
  resize3(st, bufA, bufB, (long)B*8, 24,24,24, 48,48,48, 1, tmp1, tmp2);
  k_gelu<<<GRID1D((long)B*8*110592)>>>(bufB, (long)B*8*110592);
  k_add<<<GRID1D((long)B*8*110592)>>>(bufB, t1, (long)B*8*110592);

  k_conv1x1<<<GRID1D((long)B*1*110592)>>>(bufB, fin.w, fin.b, bufA, B, 8, 1, 110592);
  float* outp = (float*)d_out;
  resize3(st, bufA, outp, (long)B*1, 48,48,48, 96,96,96, 1, tmp1, tmp2);
  k_sigmoid<<<GRID1D((long)B*884736)>>>(outp, (long)B*884736);
}